// MultiHeadedAttention_68856915690101
// MI455X (gfx1250) — compile-verified
//
#include <hip/hip_runtime.h>
#include <hip/hip_bf16.h>
#include <math.h>

// MI455X decode-MHA. Memory-bound (~1.33 GB traffic -> ~57us floor @23.3TB/s).
// fp32 WMMA (v_wmma_f32_16x16x4_f32) for all four GEMMs (exact fp32 math),
// flash-style streaming softmax for attention.

typedef float v2f __attribute__((ext_vector_type(2)));
typedef float v8f __attribute__((ext_vector_type(8)));

#define DM 4096
#define BATCH 8
#define NH 32
#define DKH 128
#define MAXL 4096

// Y(8,4096) = X(8,4096) @ W(4096,4096)^T + bias.
// One block per 16-wide output tile. 8 waves split K=4096 into 512-chunks.
// C[m][n] = sum_k A[m][k]*B[k][n]; A = X rows (m = batch), B[k][n] = W[obase+n][k].
//
// K-slot trick: within one WMMA the 4 K-slots are an arbitrary (A,B-consistent)
// set of global k's. Low half-wave (lanes 0-15, slots K0/K1) owns global k..k+3,
// high half-wave (slots K2/K3) owns k+4..k+7. One b128 load per lane then feeds
// TWO wmmas covering 8 global k's:
//   wmma0: slots = {k,k+1 | k+4,k+5},  wmma1: slots = {k+2,k+3 | k+6,k+7}.
//
// No A-row padding: garbage in A rows 8..15 only affects C rows 8..15, which
// are discarded. A loads are clamped to row (m&7) so they're always in-bounds.
__global__ __launch_bounds__(256) void gemm_tn_wmma(const float* __restrict__ X,
                                                    const float* __restrict__ W,
                                                    const float* __restrict__ Bias,
                                                    float* __restrict__ Y) {
    __shared__ float red[8][32][8];   // [wave][lane][acc-vgpr]
    const int tid  = threadIdx.x;
    const int lane = tid & 31;
    const int wave = tid >> 5;
    const int obase = blockIdx.x * 16;
    const int m    = lane & 15;               // A row (batch) / B col (output)
    const int koff = (lane < 16) ? 0 : 4;     // global-k base owned by this half-wave

    const float* __restrict__ wrow = W + (size_t)(obase + m) * DM + koff;
    const float* __restrict__ xrow = X + (size_t)(m & 7) * DM + koff;  // clamped-safe

    v8f c = {0.f, 0.f, 0.f, 0.f, 0.f, 0.f, 0.f, 0.f};

    const int k0 = wave * (DM / 8);
    const int k1 = k0 + (DM / 8);

    float4 a_cur = *reinterpret_cast<const float4*>(xrow + k0);
    float4 b_cur = *reinterpret_cast<const float4*>(wrow + k0);

    for (int k = k0; k < k1 - 8; k += 8) {
        const float4 a_nxt = *reinterpret_cast<const float4*>(xrow + k + 8);
        const float4 b_nxt = *reinterpret_cast<const float4*>(wrow + k + 8);

        v2f a0; a0.x = a_cur.x; a0.y = a_cur.y;
        v2f b0; b0.x = b_cur.x; b0.y = b_cur.y;
        c = __builtin_amdgcn_wmma_f32_16x16x4_f32(false, a0, false, b0,
                                                  (short)0, c, false, false);
        v2f a1; a1.x = a_cur.z; a1.y = a_cur.w;
        v2f b1; b1.x = b_cur.z; b1.y = b_cur.w;
        c = __builtin_amdgcn_wmma_f32_16x16x4_f32(false, a1, false, b1,
                                                  (short)0, c, false, false);
        a_cur = a_nxt;
        b_cur = b_nxt;
    }
    {   // epilogue (last 8 k's)
        v2f a0; a0.x = a_cur.x; a0.y = a_cur.y;
        v2f b0; b0.x = b_cur.x; b0.y = b_cur.y;
        c = __builtin_amdgcn_wmma_f32_16x16x4_f32(false, a0, false, b0,
                                                  (short)0, c, false, false);
        v2f a1; a1.x = a_cur.z; a1.y = a_cur.w;
        v2f b1; b1.x = b_cur.z; b1.y = b_cur.w;
        c = __builtin_amdgcn_wmma_f32_16x16x4_f32(false, a1, false, b1,
                                                  (short)0, c, false, false);
    }

    #pragma unroll
    for (int r = 0; r < 8; ++r) red[wave][lane][r] = c[r];
    __syncthreads();

    // Valid outputs: lanes 0..15 hold C rows 0..7 (batches 0..7).
    if (tid < 16) {
        const int n = tid;
        const float bb = Bias[obase + n];
        #pragma unroll
        for (int r = 0; r < 8; ++r) {
            float s = 0.f;
            #pragma unroll
            for (int w = 0; w < 8; ++w) s += red[w][n][r];
            Y[(size_t)r * DM + obase + n] = s + bb;
        }
    }
}

// One block per (b,h). 8 waves each flash-scan a chunk of the L keys.
// Lane owns dims 4*lane..4*lane+3 (one float4 = one coalesced 512B row per wave/key).
// Key/value at index cache_len come from k_new/v_new (caches are inputs: never written).
__global__ __launch_bounds__(256) void attn_kernel(const float* __restrict__ q,
                                                   const float* __restrict__ kn,
                                                   const float* __restrict__ vn,
                                                   const float* __restrict__ kcache,
                                                   const float* __restrict__ vcache,
                                                   const int* __restrict__ cache_len_p,
                                                   float* __restrict__ xout) {
    const int bh = blockIdx.x;
    const int b = bh >> 5;
    const int h = bh & 31;
    const int tid = threadIdx.x;
    const int lane = tid & 31;
    const int wave = tid >> 5;

    const int cl = *cache_len_p;
    const int L = cl + 1;
    const float scale = 0.0883883476483184f;  // 1/sqrt(128)

    const size_t headoff = (size_t)b * DM + (size_t)h * DKH;
    const float4 q4 = *reinterpret_cast<const float4*>(q + headoff + lane * 4);
    const float* __restrict__ kbase = kcache + ((size_t)(b * NH + h)) * MAXL * DKH;
    const float* __restrict__ vbase = vcache + ((size_t)(b * NH + h)) * MAXL * DKH;

    const int chunk = (L + 7) >> 3;
    const int j0 = wave * chunk;
    const int j1 = (j0 + chunk < L) ? (j0 + chunk) : L;

    float m = -3.0e38f;
    float s = 0.f;
    float4 acc = {0.f, 0.f, 0.f, 0.f};

    // Cached keys strictly below cl.
    const int jc = (j1 < cl) ? j1 : cl;
    for (int j = j0; j < jc; ++j) {
        const float4 k4 = *reinterpret_cast<const float4*>(kbase + (size_t)j * DKH + lane * 4);
        float p = q4.x * k4.x + q4.y * k4.y + q4.z * k4.z + q4.w * k4.w;
        #pragma unroll
        for (int off = 16; off; off >>= 1) p += __shfl_xor(p, off, 32);
        p *= scale;

        const float nm = fmaxf(m, p);
        const float f = expf(m - nm);
        const float e = expf(p - nm);
        const float4 v4 = *reinterpret_cast<const float4*>(vbase + (size_t)j * DKH + lane * 4);
        s = s * f + e;
        acc.x = acc.x * f + e * v4.x;
        acc.y = acc.y * f + e * v4.y;
        acc.z = acc.z * f + e * v4.z;
        acc.w = acc.w * f + e * v4.w;
        m = nm;
    }
    // Newly projected key/value at index cl (handled by the wave owning it).
    if (cl >= j0 && cl < j1) {
        const float4 k4 = *reinterpret_cast<const float4*>(kn + headoff + lane * 4);
        float p = q4.x * k4.x + q4.y * k4.y + q4.z * k4.z + q4.w * k4.w;
        #pragma unroll
        for (int off = 16; off; off >>= 1) p += __shfl_xor(p, off, 32);
        p *= scale;

        const float nm = fmaxf(m, p);
        const float f = expf(m - nm);
        const float e = expf(p - nm);
        const float4 v4 = *reinterpret_cast<const float4*>(vn + headoff + lane * 4);
        s = s * f + e;
        acc.x = acc.x * f + e * v4.x;
        acc.y = acc.y * f + e * v4.y;
        acc.z = acc.z * f + e * v4.z;
        acc.w = acc.w * f + e * v4.w;
        m = nm;
    }

    __shared__ float sm[8];
    __shared__ float ss[8];
    __shared__ float sacc[8 * DKH];
    if (lane == 0) { sm[wave] = m; ss[wave] = s; }
    sacc[wave * DKH + lane * 4 + 0] = acc.x;
    sacc[wave * DKH + lane * 4 + 1] = acc.y;
    sacc[wave * DKH + lane * 4 + 2] = acc.z;
    sacc[wave * DKH + lane * 4 + 3] = acc.w;
    __syncthreads();

    if (tid < DKH) {
        const int d = tid;
        float M = -3.0e38f;
        #pragma unroll
        for (int w = 0; w < 8; ++w) M = fmaxf(M, sm[w]);
        float S = 0.f, o = 0.f;
        #pragma unroll
        for (int w = 0; w < 8; ++w) {
            const float f = (ss[w] > 0.f) ? expf(sm[w] - M) : 0.f;
            S += ss[w] * f;
            o += sacc[w * DKH + d] * f;
        }
        xout[headoff + d] = o / S;
    }
}

extern "C" void kernel_launch(void* const* d_in, const int* in_sizes, int n_in,
                              void* d_out, int out_size, void* d_ws, size_t ws_size,
                              hipStream_t stream) {
    const float* query   = (const float*)d_in[0];
    const float* key     = (const float*)d_in[1];
    const float* value   = (const float*)d_in[2];
    const float* kcache  = (const float*)d_in[3];
    const float* vcache  = (const float*)d_in[4];
    const float* Wq      = (const float*)d_in[5];
    const float* bq      = (const float*)d_in[6];
    const float* Wk      = (const float*)d_in[7];
    const float* bk      = (const float*)d_in[8];
    const float* Wv      = (const float*)d_in[9];
    const float* bv      = (const float*)d_in[10];
    const float* Wo      = (const float*)d_in[11];
    const float* bo      = (const float*)d_in[12];
    const int*   clen    = (const int*)d_in[13];
    float* out = (float*)d_out;

    float* ws   = (float*)d_ws;
    float* qbuf = ws;                      // 8*4096
    float* knb  = ws + (size_t)BATCH * DM; // 8*4096
    float* vnb  = ws + (size_t)2 * BATCH * DM;
    float* xbuf = ws + (size_t)3 * BATCH * DM;

    const int tiles = DM / 16;  // 256
    gemm_tn_wmma<<<tiles, 256, 0, stream>>>(query, Wq, bq, qbuf);
    gemm_tn_wmma<<<tiles, 256, 0, stream>>>(key,   Wk, bk, knb);
    gemm_tn_wmma<<<tiles, 256, 0, stream>>>(value, Wv, bv, vnb);
    attn_kernel<<<BATCH * NH, 256, 0, stream>>>(qbuf, knb, vnb, kcache, vcache, clen, xbuf);
    gemm_tn_wmma<<<tiles, 256, 0, stream>>>(xbuf, Wo, bo, out);
}